// gaussian_kernel_5549097746752
// MI455X (gfx1250) — compile-verified
//
#include <hip/hip_runtime.h>

typedef float v2f __attribute__((ext_vector_type(2)));
typedef float v8f __attribute__((ext_vector_type(8)));

#define BATCH 4
#define NSRC 8192
#define SDST 2048
#define TILES_PER_BATCH (NSRC / 16)   // 512 row-tiles per batch
#define SCHUNKS (SDST / 16)           // 128 column chunks
#define WAVES_PER_BLOCK 8

// exp(-0.5*d) = exp2(d * -0.5*log2(e)) ; exp(t) = exp2(t * log2(e))
#define NEG_HALF_LOG2E (-0.72134752044448170f)
#define LOG2E          ( 1.44269504088896340f)

// One wave32 handles a 16-src-row tile for one batch.
// Cross term src·dstT via V_WMMA_F32_16X16X4_F32 (K=3 padded to 4).
// A tile (16x4):  lane L -> row M = L&15 ; half = L>>4 selects K={0,1} or {2,3}
// B tile (4x16):  lane L -> col N = L&15 ; same half/K layout
// C tile (16x16): VGPR r, lane L -> row M = r + 8*(L>>4), col N = L&15
__global__ __launch_bounds__(256) void gauss_softmax_wmma(
    const float* __restrict__ src, const float* __restrict__ dst,
    float* __restrict__ out)
{
    const int lane = threadIdx.x & 31;
    const int wid  = threadIdx.x >> 5;
    const int tile = blockIdx.x * WAVES_PER_BLOCK + wid;   // 0..2047
    const int b    = tile / TILES_PER_BATCH;
    const int m0   = (tile % TILES_PER_BATCH) << 4;
    const int hf   = lane >> 4;        // half-wave id (K-pair select)
    const int idx  = lane & 15;        // row (A) / column (B,C) index

    const float* srcB = src + (size_t)b * NSRC * 3;
    const float* dstB = dst + (size_t)b * SDST * 3;
    float*       outB = out + (size_t)b * NSRC * SDST;

    // ---- load this lane's src point (row m0+idx), build A regs ----
    const float* sp = srcB + (size_t)(m0 + idx) * 3;
    const float sx = sp[0], sy = sp[1], sz = sp[2];
    const float ssq_own = sx * sx + sy * sy + sz * sz;

    v2f A;
    A.x = hf ? sz : sx;                // K = 2*hf
    A.y = hf ? 0.0f : sy;              // K = 2*hf+1 (pad K=3 with 0)

    // src row norms for the 8 rows this lane sees in C: M = 8*hf + r.
    // ssq(m0+i) lives in lane i (0..15); gather via shuffle (loop invariant).
    float ssq[8];
    #pragma unroll
    for (int r = 0; r < 8; ++r)
        ssq[r] = __shfl(ssq_own, (hf << 3) + r, 32);

    float acc[8];
    #pragma unroll
    for (int r = 0; r < 8; ++r) acc[r] = 0.0f;

    // ================= Pass A: row sums of exp(t) =================
    // t = exp(-0.5*dist) in (0,1] -> exp(t) in [1,e]; softmax needs no
    // max-subtraction for stability here (identical math to reference).
    for (int j = 0; j < SCHUNKS; ++j) {
        const int s0 = j << 4;
        const float* dp = dstB + (size_t)(s0 + idx) * 3;
        const float dx = dp[0], dy = dp[1], dz = dp[2];
        const float dsq = dx * dx + dy * dy + dz * dz;

        v2f Bm;
        Bm.x = hf ? dz : dx;
        Bm.y = hf ? 0.0f : dy;

        v8f C = {0.f, 0.f, 0.f, 0.f, 0.f, 0.f, 0.f, 0.f};
        C = __builtin_amdgcn_wmma_f32_16x16x4_f32(
                false, A, false, Bm, (short)0, C, false, false);

        #pragma unroll
        for (int r = 0; r < 8; ++r) {
            float d2 = ssq[r] + dsq - 2.0f * C[r];
            d2 = fmaxf(d2, 1e-12f);
            float dist = __builtin_amdgcn_sqrtf(d2);
            float t = __builtin_amdgcn_exp2f(dist * NEG_HALF_LOG2E); // exp(-0.5*dist)
            acc[r] += __builtin_amdgcn_exp2f(t * LOG2E);             // exp(t)
        }
    }

    // Reduce row sums across each 16-lane group (XOR stays in-group),
    // then invert. Lane then holds inv-denominator for rows M=8*hf+r.
    #pragma unroll
    for (int r = 0; r < 8; ++r) {
        #pragma unroll
        for (int off = 8; off >= 1; off >>= 1)
            acc[r] += __shfl_xor(acc[r], off, 32);
        acc[r] = 1.0f / acc[r];
    }

    // ================= Pass B: recompute + normalized store =================
    for (int j = 0; j < SCHUNKS; ++j) {
        const int s0 = j << 4;
        const float* dp = dstB + (size_t)(s0 + idx) * 3;
        const float dx = dp[0], dy = dp[1], dz = dp[2];
        const float dsq = dx * dx + dy * dy + dz * dz;

        v2f Bm;
        Bm.x = hf ? dz : dx;
        Bm.y = hf ? 0.0f : dy;

        v8f C = {0.f, 0.f, 0.f, 0.f, 0.f, 0.f, 0.f, 0.f};
        C = __builtin_amdgcn_wmma_f32_16x16x4_f32(
                false, A, false, Bm, (short)0, C, false, false);

        #pragma unroll
        for (int r = 0; r < 8; ++r) {
            float d2 = ssq[r] + dsq - 2.0f * C[r];
            d2 = fmaxf(d2, 1e-12f);
            float dist = __builtin_amdgcn_sqrtf(d2);
            float t = __builtin_amdgcn_exp2f(dist * NEG_HALF_LOG2E);
            float o = __builtin_amdgcn_exp2f(t * LOG2E) * acc[r];
            // out[b][m0 + 8*hf + r][s0 + idx], streamed (never re-read)
            __builtin_nontemporal_store(
                o, outB + (size_t)(m0 + (hf << 3) + r) * SDST + s0 + idx);
        }
    }
}

extern "C" void kernel_launch(void* const* d_in, const int* in_sizes, int n_in,
                              void* d_out, int out_size, void* d_ws, size_t ws_size,
                              hipStream_t stream) {
    const float* src = (const float*)d_in[0];   // [4, 8192, 3] f32
    const float* dst = (const float*)d_in[1];   // [4, 2048, 3] f32
    float* out = (float*)d_out;                 // [4, 8192, 2048] f32

    const int total_tiles = BATCH * TILES_PER_BATCH;        // 2048 waves
    dim3 grid(total_tiles / WAVES_PER_BLOCK);               // 256 blocks
    dim3 block(WAVES_PER_BLOCK * 32);                       // 256 threads
    hipLaunchKernelGGL(gauss_softmax_wmma, grid, block, 0, stream,
                       src, dst, out);
}